// DynamicPositiveMask_63118839382075
// MI455X (gfx1250) — compile-verified
//
#include <hip/hip_runtime.h>
#include <math.h>

#define B_ 4
#define M_ 128
#define N_ 8192
#define C_ 3
#define CODE_ 7

typedef unsigned int u32x4 __attribute__((ext_vector_type(4)));
typedef int i32x8 __attribute__((ext_vector_type(8)));
typedef int i32x4 __attribute__((ext_vector_type(4)));

__device__ __forceinline__ float ov1(float c1, float e1, float c2, float e2) {
    float hi = fminf(c1 + 0.5f * e1, c2 + 0.5f * e2);
    float lo = fmaxf(c1 - 0.5f * e1, c2 - 0.5f * e2);
    return fmaxf(hi - lo, 0.0f);
}
__device__ __forceinline__ float en1(float c1, float e1, float c2, float e2) {
    float hi = fmaxf(c1 + 0.5f * e1, c2 + 0.5f * e2);
    float lo = fminf(c1 - 0.5f * e1, c2 - 0.5f * e2);
    return fmaxf(hi - lo, 0.0f);
}

// ---------------- Stage 1: per-element all_cost -> d_out (reused as scratch) ----
__global__ void cost_kernel(const float* __restrict__ pred_cls,
                            const float* __restrict__ target_cls,
                            const float* __restrict__ pred_reg,
                            const float* __restrict__ gt_reg,
                            const float* __restrict__ masks,
                            float* __restrict__ cost) {
    long long idx = (long long)blockIdx.x * blockDim.x + threadIdx.x;
    const long long total = (long long)B_ * M_ * N_;
    if (idx >= total) return;

    const float* pc = pred_cls + idx * C_;
    const float* tc = target_cls + idx * C_;
    float mx = pc[0] * tc[0];
    mx = fmaxf(mx, pc[1] * tc[1]);
    mx = fmaxf(mx, pc[2] * tc[2]);
    float cls_cost = 1.0f - mx;

    const float* b1 = pred_reg + idx * CODE_;
    const float* b2 = gt_reg + idx * CODE_;
    float a0 = b1[0], a1 = b1[1], a2 = b1[2], a3 = b1[3], a4 = b1[4], a5 = b1[5], a6 = b1[6];
    float g0 = b2[0], g1 = b2[1], g2 = b2[2], g3 = b2[3], g4 = b2[4], g5 = b2[5], g6 = b2[6];

    float isnotnan = 1.0f;
    if ((g0 != g0) || (g1 != g1) || (g2 != g2) || (g3 != g3) ||
        (g4 != g4) || (g5 != g5) || (g6 != g6)) isnotnan = 0.0f;

    const float vx = 0.8f, vy = 0.8f, J = 0.5f;
    float x1 = a0 * vx, y1 = a1 * vy, z1 = a2;
    float x2 = g0 * vx, y2 = g1 * vy, z2 = g2;
    float l1 = fminf(expf(a3), 10.0f), w1 = fminf(expf(a4), 10.0f), h1 = fminf(expf(a5), 10.0f);
    float l2 = fminf(expf(g3), 10.0f), w2 = fminf(expf(g4), 10.0f), h2 = fminf(expf(g5), 10.0f);
    float s1, c1s, s2, c2s;
    sincosf(a6, &s1, &c1s);
    sincosf(g6, &s2, &c2s);
    float t1 = s1 * c2s;
    float t2 = c1s * s2;

    float vol1 = l1 * w1 * h1 * J;
    float vol2 = l2 * w2 * h2 * J;
    float inter = ov1(x1, l1, x2, l2) * ov1(y1, w1, y2, w2) *
                  ov1(z1, h1, z2, h2) * ov1(t1, J, t2, J);
    float uni = vol1 + vol2 - inter;
    float dx = x2 - x1, dy = y2 - y1, dz = z2 - z1, dt = t2 - t1;
    float inter_diag = dx * dx + dy * dy + dz * dz + dt * dt;
    float ex = en1(x1, l1, x2, l2), ey = en1(y1, w1, y2, w2);
    float ez = en1(z1, h1, z2, h2), et = en1(t1, J, t2, J);
    float c_diag = ex * ex + ey * ey + ez * ez + et * et;
    float u = inter_diag / c_diag;
    float rdiou = inter / uni;

    float m = masks[idx];
    float reg_cost = (1.0f - fminf(fmaxf(rdiou, 0.0f), 1.0f) + u) * m * isnotnan;
    cost[idx] = cls_cost * m + 2.0f * reg_cost + (1.0f - m) * 100.0f;
}

// ---------------- Stage 2: per-row valid flag ----------------
__global__ void rowvalid_kernel(const float* __restrict__ masks, int* __restrict__ rowvalid) {
    __shared__ float red[256];
    int r = blockIdx.x;
    const float* row = masks + (long long)r * N_;
    float s = 0.0f;
    for (int n = threadIdx.x; n < N_; n += 256) s += row[n];
    red[threadIdx.x] = s;
    __syncthreads();
    for (int off = 128; off > 0; off >>= 1) {
        if (threadIdx.x < off) red[threadIdx.x] += red[threadIdx.x + off];
        __syncthreads();
    }
    if (threadIdx.x == 0) rowvalid[r] = (red[0] > 0.0f) ? 1 : 0;
}

// ---------------- Stage 3: per-batch box count ----------------
__global__ void boxnum_kernel(const int* __restrict__ rowvalid, int* __restrict__ boxnum) {
    __shared__ int red[M_];
    int b = blockIdx.x;
    red[threadIdx.x] = rowvalid[b * M_ + threadIdx.x];
    __syncthreads();
    for (int off = M_ / 2; off > 0; off >>= 1) {
        if (threadIdx.x < off) red[threadIdx.x] += red[threadIdx.x + off];
        __syncthreads();
    }
    if (threadIdx.x == 0) boxnum[b] = red[0];
}

// ---------------- Stage 4: per-row radix top-k select + final write ----------------
__global__ __launch_bounds__(1024) void select_kernel(const float* __restrict__ iou_target,
                                                      const float* __restrict__ masks,
                                                      const int* __restrict__ boxnum,
                                                      float* __restrict__ out /* holds cost */) {
    __shared__ float costLds[N_];
    __shared__ unsigned hist[256];
    __shared__ unsigned scanbuf[1024];
    __shared__ float redf[1024];
    __shared__ unsigned sh_prefix;
    __shared__ int sh_want;

    const int tid = threadIdx.x;
    const int r = blockIdx.x;
    const int b = r / M_;
    const int m = r % M_;
    const long long base = (long long)r * N_;

    // active = m < box_num[b] (uniform across block)
    if (m >= boxnum[b]) {
        float4 z = make_float4(0.0f, 0.0f, 0.0f, 0.0f);
        float4* o = (float4*)(out + base);
        o[tid * 2 + 0] = z;
        o[tid * 2 + 1] = z;
        return;
    }

    // Issue TDM load of the 8192-float cost row into LDS (one DMA per block).
    if (tid == 0) {
        unsigned lds_off = (unsigned)(unsigned long long)(void*)costLds; // low 32 bits = LDS byte offset
        unsigned long long ga = (unsigned long long)(out + base);
        u32x4 g0;
        g0.x = 1u;                                   // [1:0] count = 1 valid descriptor
        g0.y = lds_off;                              // [63:32] lds_addr
        g0.z = (unsigned)(ga & 0xFFFFFFFFull);       // [95:64] global_addr lo
        g0.w = (unsigned)((ga >> 32) & 0x1FFFFFFull) // [120:96] global_addr hi (57-bit addr)
               | (2u << 30);                         // [127:126] type = 2 ("image")
        i32x8 g1;
        g1[0] = (int)(2u << 16);                            // data_size = 4 bytes
        g1[1] = (int)(((unsigned)N_ & 0xFFFFu) << 16);      // tensor_dim0 lo16 @ [63:48]
        g1[2] = (int)((((unsigned)N_ >> 16) & 0xFFFFu) | (1u << 16)); // dim0 hi16, tensor_dim1 = 1
        g1[3] = (int)(((unsigned)N_ & 0xFFFFu) << 16);      // tile_dim0 = N_ @ [127:112]
        g1[4] = 1;                                          // tile_dim1 = 1, tile_dim2 = 0
        g1[5] = N_;                                         // tensor_dim0_stride lo32
        g1[6] = 0;
        g1[7] = 0;
        i32x4 gz4 = {0, 0, 0, 0};
        i32x8 gz8 = {0, 0, 0, 0, 0, 0, 0, 0};
        __builtin_amdgcn_tensor_load_to_lds(g0, g1, gz4, gz4, gz8, 0);
    }

    // Overlap: load iou & mask rows into registers (fully coalesced float4).
    const float4* iouv = (const float4*)(iou_target + base);
    const float4* mv = (const float4*)(masks + base);
    float4 i0 = iouv[tid * 2 + 0], i1 = iouv[tid * 2 + 1];
    float4 m0 = mv[tid * 2 + 0], m1 = mv[tid * 2 + 1];
    float iou8[8] = {i0.x, i0.y, i0.z, i0.w, i1.x, i1.y, i1.z, i1.w};
    float msk8[8] = {m0.x, m0.y, m0.z, m0.w, m1.x, m1.y, m1.z, m1.w};

    // k = trunc(max(sum(iou_row), 1))  (deterministic tree reduction)
    float ls = 0.0f;
    for (int i = 0; i < 8; ++i) ls += iou8[i];
    redf[tid] = ls;
    __syncthreads();
    for (int off = 512; off > 0; off >>= 1) {
        if (tid < off) redf[tid] += redf[tid + off];
        __syncthreads();
    }
    int k = (int)fmaxf(redf[0], 1.0f);
    if (k > N_) k = N_;

    // Wait for tensor DMA in the issuing wave, then publish to all waves.
    if (tid == 0) __builtin_amdgcn_s_wait_tensorcnt(0);
    __syncthreads();

    // Order-preserving float -> uint key map.
    unsigned keys[8];
    const int n0 = tid * 8;
    for (int i = 0; i < 8; ++i) {
        unsigned u = __float_as_uint(costLds[n0 + i]);
        keys[i] = (u & 0x80000000u) ? ~u : (u | 0x80000000u);
    }

    // 4-pass MSB radix select: find key of rank k (1-indexed).
    unsigned prefix = 0;
    int want = k;
    for (int s = 24; s >= 0; s -= 8) {
        if (tid < 256) hist[tid] = 0u;
        __syncthreads();
        for (int i = 0; i < 8; ++i) {
            unsigned key = keys[i];
            unsigned hi = key >> ((s + 8) & 31);
            bool match = (s == 24) || (hi == prefix);
            if (match) atomicAdd(&hist[(key >> s) & 255u], 1u);
        }
        __syncthreads();
        if (tid == 0) {
            unsigned running = 0;
            unsigned bucket = 255u;
            for (unsigned j = 0; j < 256u; ++j) {
                unsigned c = hist[j];
                if (running + c >= (unsigned)want) { bucket = j; break; }
                running += c;
            }
            sh_prefix = (prefix << 8) | bucket;
            sh_want = want - (int)running;
        }
        __syncthreads();
        prefix = sh_prefix;
        want = sh_want;
        __syncthreads();
    }
    const unsigned T = prefix;   // exact k-th smallest key
    const int need_eq = want;    // how many key==T elements to take (smallest index first)

    // Stable tie-break: block-wide exclusive scan of per-thread equal counts
    // (threads own index-contiguous chunks, so scan order == index order).
    int cnt = 0;
    for (int i = 0; i < 8; ++i)
        if (keys[i] == T) cnt++;
    scanbuf[tid] = (unsigned)cnt;
    __syncthreads();
    for (int off = 1; off < 1024; off <<= 1) {
        unsigned v = scanbuf[tid];
        unsigned a = (tid >= off) ? scanbuf[tid - off] : 0u;
        __syncthreads();
        scanbuf[tid] = v + a;
        __syncthreads();
    }
    int ex = (int)scanbuf[tid] - cnt;

    float res[8];
    for (int i = 0; i < 8; ++i) {
        unsigned key = keys[i];
        bool sel;
        if (key < T) {
            sel = true;
        } else if (key == T) {
            sel = (ex < need_eq);
            ex++;
        } else {
            sel = false;
        }
        float pos = sel ? 1.0f : iou8[i];
        res[i] = pos * msk8[i];
    }
    float4* o = (float4*)(out + base);
    o[tid * 2 + 0] = make_float4(res[0], res[1], res[2], res[3]);
    o[tid * 2 + 1] = make_float4(res[4], res[5], res[6], res[7]);
}

extern "C" void kernel_launch(void* const* d_in, const int* in_sizes, int n_in,
                              void* d_out, int out_size, void* d_ws, size_t ws_size,
                              hipStream_t stream) {
    const float* pred_cls   = (const float*)d_in[0];
    const float* target_cls = (const float*)d_in[1];
    const float* pred_reg   = (const float*)d_in[2];
    const float* gt_reg     = (const float*)d_in[3];
    const float* masks      = (const float*)d_in[4];
    const float* iou_target = (const float*)d_in[5];
    float* out = (float*)d_out;

    int* rowvalid = (int*)d_ws;          // B_*M_ ints
    int* boxnum = rowvalid + B_ * M_;    // B_ ints

    const long long total = (long long)B_ * M_ * N_;
    const int blocks = (int)((total + 255) / 256);

    cost_kernel<<<blocks, 256, 0, stream>>>(pred_cls, target_cls, pred_reg, gt_reg, masks, out);
    rowvalid_kernel<<<B_ * M_, 256, 0, stream>>>(masks, rowvalid);
    boxnum_kernel<<<B_, M_, 0, stream>>>(rowvalid, boxnum);
    select_kernel<<<B_ * M_, 1024, 0, stream>>>(iou_target, masks, boxnum, out);
}